// DParser_55113020342733
// MI455X (gfx1250) — compile-verified
//
#include <hip/hip_runtime.h>

#define N_ 1024
#define D_ 768
#define H_ 128

typedef __attribute__((ext_vector_type(2))) float v2f;
typedef __attribute__((ext_vector_type(8))) float v8f;

// ---------------------------------------------------------------------------
// Kernel 1: front GEMMs on the WMMA pipe (fp32, 16x16x4).
//   sel==0:  A  = hidden @ W1[:768]
//   sel==1:  B' = hidden @ W1[768:] + b1      (b1 folded in here)
// Each wave32 owns TWO 16x16 output tiles stacked in M (32 rows x 16 cols),
// sharing one B fragment per k-step: 2 WMMAs per W-load pair (halves W
// traffic vs 1 tile/wave) and two independent accumulator chains for the
// tensor pipe.  512 waves total.
// ---------------------------------------------------------------------------
__global__ void __launch_bounds__(128)
gemm_front(const float* __restrict__ hidden,
           const float* __restrict__ W1,
           const float* __restrict__ b1,
           float* __restrict__ outA,
           float* __restrict__ outB)
{
    const int lane  = threadIdx.x & 31;
    const int wave  = threadIdx.x >> 5;
    const int ntile = blockIdx.x * 4 + wave;   // 0..7   (H/16)
    const int mbase = blockIdx.y * 32;         // 0..992 (two 16-row tiles)
    const int sel   = blockIdx.z;              // 0 -> A, 1 -> B

    const float* W  = W1 + (size_t)sel * D_ * H_;
    float* outM     = sel ? outB : outA;

    const int half = lane >> 4;                // 0 or 1
    const int m    = lane & 15;                // A-frag row / B-frag column
    const int col  = ntile * 16 + m;           // output column
    const int koff = half * 2;                 // K sub-offset per lane half

    v8f c0 = {0.f, 0.f, 0.f, 0.f, 0.f, 0.f, 0.f, 0.f};
    v8f c1 = {0.f, 0.f, 0.f, 0.f, 0.f, 0.f, 0.f, 0.f};

    const float* aptr0 = hidden + (size_t)(mbase + m)      * D_ + koff;
    const float* aptr1 = hidden + (size_t)(mbase + 16 + m) * D_ + koff;

    #pragma unroll 4
    for (int k0 = 0; k0 < D_; k0 += 4) {
        // Shared B fragment: lane holds W[k0+koff, col], W[k0+koff+1, col]
        v2f b;
        b.x = W[(size_t)(k0 + koff)     * H_ + col];
        b.y = W[(size_t)(k0 + koff + 1) * H_ + col];
        // Two A fragments (rows mbase.. and mbase+16..), 8B loads each
        v2f a0 = *(const v2f*)(aptr0 + k0);
        v2f a1 = *(const v2f*)(aptr1 + k0);
        c0 = __builtin_amdgcn_wmma_f32_16x16x4_f32(false, a0, false, b,
                                                   (short)0, c0, false, false);
        c1 = __builtin_amdgcn_wmma_f32_16x16x4_f32(false, a1, false, b,
                                                   (short)0, c1, false, false);
    }

    const float badd = sel ? b1[col] : 0.f;
    #pragma unroll
    for (int v = 0; v < 8; ++v) {
        const int r = v + half * 8;            // C/D layout row within tile
        outM[(size_t)(mbase + r)      * H_ + col] = c0[v] + badd;
        outM[(size_t)(mbase + 16 + r) * H_ + col] = c1[v] + badd;
    }
}

// ---------------------------------------------------------------------------
// Kernel 2: fused pairwise relu-dot (never materializes the 512MB act tensor).
// Block = 16 i x 64 j; each thread carries 4 accumulators (j, j+16, j+32,
// j+48) so every sA/sW LDS read feeds 4 FMA chains (dual-issue friendly).
//   scores[i,j] = sum_h relu(A[i,h] + B'[j,h]) * W2[h] + b2
//   out[0, j]   = scores[j, j]
//   out[1+i, j] = (i==j) ? -inf : scores[i, j]
// ---------------------------------------------------------------------------
__global__ void __launch_bounds__(256)
pairwise_score(const float* __restrict__ A,
               const float* __restrict__ Bp,
               const float* __restrict__ W2,
               const float* __restrict__ b2,
               float* __restrict__ out)
{
    __shared__ float sA[16][132];   // pad 128->132: bank step 4, conflict-free
    __shared__ float sB[64][132];
    __shared__ float sW[128];

    const int tx  = threadIdx.x;            // j lane in tile
    const int ty  = threadIdx.y;            // i in tile
    const int tid = ty * 16 + tx;
    const int j0  = blockIdx.x * 64;
    const int i0  = blockIdx.y * 16;

    // Stage A tile (16x128 = 512 float4) and B tile (64x128 = 2048 float4).
    #pragma unroll
    for (int it = 0; it < 2; ++it) {
        const int idx = tid + it * 256;
        const int r   = idx >> 5;
        const int c4  = (idx & 31) << 2;
        *(float4*)&sA[r][c4] = *(const float4*)(A + (size_t)(i0 + r) * H_ + c4);
    }
    #pragma unroll
    for (int it = 0; it < 8; ++it) {
        const int idx = tid + it * 256;
        const int r   = idx >> 5;
        const int c4  = (idx & 31) << 2;
        *(float4*)&sB[r][c4] = *(const float4*)(Bp + (size_t)(j0 + r) * H_ + c4);
    }
    if (tid < 128) sW[tid] = W2[tid];
    const float b2v = b2[0];
    __syncthreads();

    float acc0 = 0.f, acc1 = 0.f, acc2 = 0.f, acc3 = 0.f;
    #pragma unroll 4
    for (int h = 0; h < H_; h += 4) {
        const float4 a4 = *(const float4*)&sA[ty][h];
        const float4 w4 = *(const float4*)&sW[h];
        const float4 b0 = *(const float4*)&sB[tx][h];
        const float4 b1v = *(const float4*)&sB[tx + 16][h];
        const float4 b2q = *(const float4*)&sB[tx + 32][h];
        const float4 b3 = *(const float4*)&sB[tx + 48][h];

        acc0 += fmaxf(a4.x + b0.x, 0.f) * w4.x;
        acc1 += fmaxf(a4.x + b1v.x, 0.f) * w4.x;
        acc2 += fmaxf(a4.x + b2q.x, 0.f) * w4.x;
        acc3 += fmaxf(a4.x + b3.x, 0.f) * w4.x;

        acc0 += fmaxf(a4.y + b0.y, 0.f) * w4.y;
        acc1 += fmaxf(a4.y + b1v.y, 0.f) * w4.y;
        acc2 += fmaxf(a4.y + b2q.y, 0.f) * w4.y;
        acc3 += fmaxf(a4.y + b3.y, 0.f) * w4.y;

        acc0 += fmaxf(a4.z + b0.z, 0.f) * w4.z;
        acc1 += fmaxf(a4.z + b1v.z, 0.f) * w4.z;
        acc2 += fmaxf(a4.z + b2q.z, 0.f) * w4.z;
        acc3 += fmaxf(a4.z + b3.z, 0.f) * w4.z;

        acc0 += fmaxf(a4.w + b0.w, 0.f) * w4.w;
        acc1 += fmaxf(a4.w + b1v.w, 0.f) * w4.w;
        acc2 += fmaxf(a4.w + b2q.w, 0.f) * w4.w;
        acc3 += fmaxf(a4.w + b3.w, 0.f) * w4.w;
    }

    const int   i = i0 + ty;
    const float acc[4] = {acc0, acc1, acc2, acc3};
    #pragma unroll
    for (int q = 0; q < 4; ++q) {
        const int   j     = j0 + tx + q * 16;
        const float score = acc[q] + b2v;
        out[(size_t)(1 + i) * N_ + j] = (i == j) ? -__builtin_inff() : score;
        if (i == j) out[j] = score;   // diag row (pre-mask)
    }
}

// ---------------------------------------------------------------------------
extern "C" void kernel_launch(void* const* d_in, const int* in_sizes, int n_in,
                              void* d_out, int out_size, void* d_ws, size_t ws_size,
                              hipStream_t stream) {
    const float* hidden = (const float*)d_in[0];   // (1024, 768)
    const float* W1     = (const float*)d_in[1];   // (1536, 128)
    const float* b1     = (const float*)d_in[2];   // (128,)
    const float* W2     = (const float*)d_in[3];   // (128,)
    const float* b2     = (const float*)d_in[4];   // (1,)
    float* out = (float*)d_out;                    // (1025, 1024)

    float* A  = (float*)d_ws;                      // 1024x128
    float* Bp = A + (size_t)N_ * H_;               // 1024x128 (with b1 folded)

    // 2 x 32 x 2 blocks x 4 waves: one wave per 32x16 slab of A / B'.
    gemm_front<<<dim3(2, 32, 2), dim3(128), 0, stream>>>(hidden, W1, b1, A, Bp);

    // 16 x 64 blocks: each covers a 16(i) x 64(j) slab of the score matrix.
    pairwise_score<<<dim3(16, 64), dim3(16, 16), 0, stream>>>(A, Bp, W2, b2, out);
}